// Post_process_deconv_35570919146245
// MI455X (gfx1250) — compile-verified
//
#include <hip/hip_runtime.h>
#include <hip/hip_bf16.h>

// Deformable 3x3 post-process:
//   out = depth + b + sum_k w_k * (weight_k - mean_c weight_c) * bilinear(depth, y-1+kh+dy_k, x-1+kw+dx_k)
//
// Memory-bound (~99 MB HBM traffic -> ~4.3 us floor @ 23.3 TB/s). Strategy:
//  - 4 pixels/thread -> all weight/offset channel reads are coalesced global_load_b128, NT hint
//  - depth kept at default (RT) temporal policy so its 3.4 MB stays hot in the 192 MB L2;
//    all bilinear gathers are L2 hits
//  - x-adjacent bilinear corner pairs fetched as one 8B load (18 gathers/px instead of 36),
//    addresses clamped in-bounds, validity applied branchlessly
//  - wave32: 256-thread blocks (8 waves)

#define HN 352
#define WN 1216
#define BN 2
#define WQ (WN / 4)              // 304 quads per row
#define HW (HN * WN)             // 428032 elements per image plane

typedef float v4f __attribute__((ext_vector_type(4)));   // true clang vector: NT-builtin compatible

struct F2 { float a, b; };

__device__ __forceinline__ float bilin(const float* __restrict__ img, float py, float px) {
    float fy = floorf(py);
    float fx = floorf(px);
    float ly = py - fy;
    float lx = px - fx;
    int y0 = (int)fy;
    int x0 = (int)fx;
    int y1 = y0 + 1;
    int x1 = x0 + 1;

    bool vy0 = (unsigned)y0 < (unsigned)HN;
    bool vy1 = (unsigned)y1 < (unsigned)HN;
    bool vx0 = (unsigned)x0 < (unsigned)WN;
    bool vx1 = (unsigned)x1 < (unsigned)WN;

    // Clamp addresses in-bounds; fetch each row's corner pair as one 8-byte load.
    int yc0 = min(max(y0, 0), HN - 1);
    int yc1 = min(max(y1, 0), HN - 1);
    int xc  = min(max(x0, 0), WN - 2);

    F2 r0, r1;
    __builtin_memcpy(&r0, img + yc0 * WN + xc, 8);
    __builtin_memcpy(&r1, img + yc1 * WN + xc, 8);

    // If x0 == xc the pair is (x0, x0+1); otherwise it is shifted by one
    // (x0 == -1 -> pair starts at 0; x0 == WN-1 -> pair starts at WN-2).
    bool lo = (x0 == xc);
    float a00 = lo ? r0.a : r0.b;
    float a01 = lo ? r0.b : r0.a;
    float a10 = lo ? r1.a : r1.b;
    float a11 = lo ? r1.b : r1.a;

    float v00 = (vy0 && vx0) ? a00 : 0.0f;
    float v01 = (vy0 && vx1) ? a01 : 0.0f;
    float v10 = (vy1 && vx0) ? a10 : 0.0f;
    float v11 = (vy1 && vx1) ? a11 : 0.0f;

    return (1.0f - ly) * ((1.0f - lx) * v00 + lx * v01) +
           ly * ((1.0f - lx) * v10 + lx * v11);
}

__global__ __launch_bounds__(256)
void Post_process_deconv_35570919146245_kernel(
    const float* __restrict__ depth,    // [B,1,H,W]
    const float* __restrict__ weight,   // [B,9,H,W]
    const float* __restrict__ offset,   // [B,18,H,W]
    const float* __restrict__ wk9,      // [9]  (w, row-major kh,kw)
    const float* __restrict__ bias,     // [1]
    float* __restrict__ out,            // [B,1,H,W]
    int total)                          // B*H*(W/4)
{
    int tid = blockIdx.x * blockDim.x + threadIdx.x;
    if (tid >= total) return;

    int xq = tid % WQ;
    int t  = tid / WQ;
    int y  = t % HN;
    int bb = t / HN;
    int x0 = xq * 4;

    const float* dimg = depth + bb * HW;
    int rowoff = y * WN + x0;

    // Streamed center-depth quad (RT: keep depth resident in L2 for the gathers).
    v4f d4 = *reinterpret_cast<const v4f*>(dimg + rowoff);

    float bv = bias[0];

    // out_j = accws_j - (wsum_j/9) * accs_j + bv + depth_j
    //   accws = sum_k w_k * weight_k * s_k
    //   accs  = sum_k w_k * s_k
    //   wsum  = sum_k weight_k
    float accws[4] = {0.f, 0.f, 0.f, 0.f};
    float accs[4]  = {0.f, 0.f, 0.f, 0.f};
    float wsum[4]  = {0.f, 0.f, 0.f, 0.f};

    const float* wbase = weight + bb * 9 * HW + rowoff;
    const float* obase = offset + bb * 18 * HW + rowoff;

#pragma unroll
    for (int k = 0; k < 9; ++k) {
        int kh = k / 3;
        int kw = k % 3;
        float wk = wk9[k];  // uniform scalar load

        // One-touch streams: non-temporal b128 loads.
        v4f wv = __builtin_nontemporal_load(
            reinterpret_cast<const v4f*>(wbase + k * HW));
        v4f dy = __builtin_nontemporal_load(
            reinterpret_cast<const v4f*>(obase + (2 * k) * HW));
        v4f dx = __builtin_nontemporal_load(
            reinterpret_cast<const v4f*>(obase + (2 * k + 1) * HW));

        float pyb = (float)(y - 1 + kh);
        float pxb = (float)(x0 - 1 + kw);

#pragma unroll
        for (int j = 0; j < 4; ++j) {
            float s = bilin(dimg, pyb + dy[j], pxb + (float)j + dx[j]);
            accws[j] = fmaf(wk * wv[j], s, accws[j]);
            accs[j]  = fmaf(wk, s, accs[j]);
            wsum[j] += wv[j];
        }
    }

    v4f o;
#pragma unroll
    for (int j = 0; j < 4; ++j) {
        o[j] = accws[j] - (wsum[j] * (1.0f / 9.0f)) * accs[j] + bv + d4[j];
    }

    __builtin_nontemporal_store(o, reinterpret_cast<v4f*>(out + bb * HW + rowoff));
}

extern "C" void kernel_launch(void* const* d_in, const int* in_sizes, int n_in,
                              void* d_out, int out_size, void* d_ws, size_t ws_size,
                              hipStream_t stream) {
    (void)in_sizes; (void)n_in; (void)d_ws; (void)ws_size; (void)out_size;

    const float* depth  = (const float*)d_in[0];
    const float* weight = (const float*)d_in[1];
    const float* offset = (const float*)d_in[2];
    const float* w      = (const float*)d_in[3];
    const float* b      = (const float*)d_in[4];
    float* out          = (float*)d_out;

    int total = BN * HN * WQ;            // 214016 threads, 4 px each
    int block = 256;                     // 8 wave32 per block
    int grid  = (total + block - 1) / block;

    Post_process_deconv_35570919146245_kernel<<<grid, block, 0, stream>>>(
        depth, weight, offset, w, b, out, total);
}